// LddtLoss_89292370084243
// MI455X (gfx1250) — compile-verified
//
#include <hip/hip_runtime.h>

typedef float v2f __attribute__((ext_vector_type(2)));
typedef float v8f __attribute__((ext_vector_type(8)));

#define B_DIM   8
#define N_DIM   2048
#define TILES   (N_DIM / 16)      // 128 column tiles of 16
#define STRIPES (N_DIM / 16)      // 128 row stripes of 16
#define CUT2    225.0f            // 15.0^2
#define EPS2    1e-16f            // (1e-8)^2
#define EPS_F   1e-8f

// One block = one (batch, 16-row stripe). 8 waves; wave w handles column
// tiles jt = stripe+w, stripe+w+8, ... (upper triangle only; off-diagonal
// tiles weighted 2x by symmetry of both distance matrices).
// Each tile: two v_wmma_f32_16x16x4_f32 (pred & true Gram, K padded 3->4),
// then a branch-free lDDT epilogue with ONE raw v_sqrt_f32 per pair:
//   |sqrt(a)-sqrt(b)|^2 = a + b - 2*sqrt(a*b), compared to squared bins.
// Column norms are computed per-lane from the already-loaded b96 triple
// (no cross-lane shuffle, no DS traffic in the loop).
__global__ __launch_bounds__(256) void lddt_tiles_kernel(
    const float* __restrict__ pred, const float* __restrict__ tru,
    float* __restrict__ partial)
{
    __shared__ float rowp[16];   // |pred row|^2 for the 16 rows of this stripe
    __shared__ float rowt[16];   // |true row|^2
    __shared__ float red[16];    // cross-wave reduction buffer

    const int tid  = threadIdx.x;
    const int wave = tid >> 5;
    const int lane = tid & 31;
    const int b    = blockIdx.x >> 7;                    // batch
    const int st   = blockIdx.x & (STRIPES - 1);         // stripe index
    const int i0   = st << 4;                            // first row of stripe

    const float* __restrict__ pb = pred + (size_t)b * N_DIM * 3;
    const float* __restrict__ tb = tru  + (size_t)b * N_DIM * 3;

    // Row norms, once per block.
    if (tid < 16) {
        const float* p = pb + (size_t)(i0 + tid) * 3;
        rowp[tid] = p[0] * p[0] + p[1] * p[1] + p[2] * p[2];
    } else if (tid < 32) {
        const float* t = tb + (size_t)(i0 + (tid - 16)) * 3;
        rowt[tid - 16] = t[0] * t[0] + t[1] * t[1] + t[2] * t[2];
    }
    __syncthreads();

    // A fragment (rows i0..i0+15). ISA layout for 16x4 f32 A:
    // lanes 0-15: VGPR0=K0, VGPR1=K1 (x,y); lanes 16-31: VGPR0=K2, VGPR1=K3 (z,0).
    const int hi = lane >> 4;       // which K-half this lane holds
    const int rm = lane & 15;       // row (for A) / column (for B/C/D) index
    v2f ap, at;
    {
        const float* p = pb + (size_t)(i0 + rm) * 3;
        const float* t = tb + (size_t)(i0 + rm) * 3;
        float px = p[0], py = p[1], pz = p[2];
        float tx = t[0], ty = t[1], tz = t[2];
        ap.x = hi ? pz : px;  ap.y = hi ? 0.0f : py;   // K pad (K=3) = 0
        at.x = hi ? tz : tx;  at.y = hi ? 0.0f : ty;
    }

    float num = 0.0f, den = 0.0f;

    for (int jt = st + wave; jt < TILES; jt += 8) {
        const int c = (jt << 4) + rm;   // this lane's column
        const float* p = pb + (size_t)c * 3;
        const float* t = tb + (size_t)c * 3;
        float px = p[0], py = p[1], pz = p[2];
        float tx = t[0], ty = t[1], tz = t[2];

        // B fragment (4x16): mirrors A layout across the half-waves.
        v2f bp, bt;
        bp.x = hi ? pz : px;  bp.y = hi ? 0.0f : py;
        bt.x = hi ? tz : tx;  bt.y = hi ? 0.0f : ty;

        // Full column norms straight from the loaded triples — every lane
        // has all three components in registers; no cross-lane exchange.
        float colp = px * px + py * py + pz * pz;
        float colt = tx * tx + ty * ty + tz * tz;

        v8f accp = {}; v8f acct = {};
        accp = __builtin_amdgcn_wmma_f32_16x16x4_f32(
            false, ap, false, bp, (short)0, accp, false, false);
        acct = __builtin_amdgcn_wmma_f32_16x16x4_f32(
            false, at, false, bt, (short)0, acct, false, false);

        float tn = 0.0f, td = 0.0f;
        #pragma unroll
        for (int e = 0; e < 8; ++e) {
            // C/D layout: element e is row M = e + 8*hi, col N = lane&15.
            float rp = rowp[e + 8 * hi];
            float rt = rowt[e + 8 * hi];
            float d2p = fmaxf(rp + colp - 2.0f * accp[e], 0.0f);
            float d2t = fmaxf(rt + colt - 2.0f * acct[e], 0.0f);
            // mask straight from squared true distance (no sqrt needed)
            float m = (d2t < CUT2 && d2t > EPS2) ? 1.0f : 0.0f;
            // |dp - dt|^2 with a single raw v_sqrt_f32 (no IEEE fixup seq.)
            float s     = __builtin_amdgcn_sqrtf(d2p * d2t);
            float diff2 = d2p + d2t - 2.0f * s;
            float pres = 0.0f;
            pres += (diff2 < 0.25f) ? 0.5f   : 0.0f;   // diff < 0.5
            pres += (diff2 < 1.0f)  ? 0.25f  : 0.0f;   // diff < 1
            pres += (diff2 < 4.0f)  ? 0.125f : 0.0f;   // diff < 2
            pres += (diff2 < 16.0f) ? 0.125f : 0.0f;   // diff < 4
            tn += pres * m;
            td += m;
        }
        // Off-diagonal tiles stand in for their transposed twin as well.
        const float wgt = (jt == st) ? 1.0f : 2.0f;
        num += wgt * tn;
        den += wgt * td;
    }

    // Wave reduction (wave32), then cross-wave via LDS. Deterministic.
    #pragma unroll
    for (int off = 16; off; off >>= 1) {
        num += __shfl_down(num, off, 32);
        den += __shfl_down(den, off, 32);
    }
    if (lane == 0) { red[wave] = num; red[8 + wave] = den; }
    __syncthreads();
    if (tid == 0) {
        float n = 0.0f, d = 0.0f;
        #pragma unroll
        for (int w = 0; w < 8; ++w) { n += red[w]; d += red[8 + w]; }
        partial[(size_t)blockIdx.x * 2 + 0] = n;
        partial[(size_t)blockIdx.x * 2 + 1] = d;
    }
}

// Single block: wave w reduces the 128 stripe-partials of batch w, computes
// (1 - num/max(den,eps)); thread 0 averages the 8 batches into d_out[0].
__global__ __launch_bounds__(256) void lddt_final_kernel(
    const float* __restrict__ partial, float* __restrict__ out)
{
    __shared__ float acc[8];
    const int tid = threadIdx.x, w = tid >> 5, lane = tid & 31;
    float num = 0.0f, den = 0.0f;
    for (int i = lane; i < STRIPES; i += 32) {
        num += partial[(size_t)(w * STRIPES + i) * 2 + 0];
        den += partial[(size_t)(w * STRIPES + i) * 2 + 1];
    }
    #pragma unroll
    for (int off = 16; off; off >>= 1) {
        num += __shfl_down(num, off, 32);
        den += __shfl_down(den, off, 32);
    }
    if (lane == 0) acc[w] = 1.0f - num / fmaxf(den, EPS_F);
    __syncthreads();
    if (tid == 0) {
        float s = 0.0f;
        #pragma unroll
        for (int i = 0; i < B_DIM; ++i) s += acc[i];
        out[0] = s * (1.0f / (float)B_DIM);
    }
}

extern "C" void kernel_launch(void* const* d_in, const int* in_sizes, int n_in,
                              void* d_out, int out_size, void* d_ws, size_t ws_size,
                              hipStream_t stream) {
    const float* pred = (const float*)d_in[0];
    const float* tru  = (const float*)d_in[1];
    float* out        = (float*)d_out;
    float* partial    = (float*)d_ws;   // B_DIM*STRIPES*2 floats = 8 KB

    dim3 grid1(B_DIM * STRIPES);        // 1024 blocks
    lddt_tiles_kernel<<<grid1, 256, 0, stream>>>(pred, tru, partial);
    lddt_final_kernel<<<1, 256, 0, stream>>>(partial, out);
}